// Attn_Decoder_31164282700145
// MI455X (gfx1250) — compile-verified
//
#include <hip/hip_runtime.h>
#include <math.h>

#define D_DIM  2048
#define IN_DIM 512
#define T_DIM  4096

typedef float v2f __attribute__((ext_vector_type(2)));
typedef float v8f __attribute__((ext_vector_type(8)));

// ---------------------------------------------------------------------------
// Fused GEMM + attention-score epilogue (fp32 WMMA 16x16x4).
//   raw[t,d] = enc[t,:] . U_w[d,:]
//   score[t] += sum_d tanh(q[d] + raw*prod(t)) * V_w[d]
// WG = 256 thr = 8 waves arranged 2(M) x 4(N). WG tile 64M x 128N.
// Each wave register-blocks 32M x 32N = 2x2 accumulators of 16x16.
// LDS tiles are stored K-major ([m][k], [n][k], pitch KC+2=34 dwords) so each
// WMMA fragment ({K=kA,K=kA+1} at fixed column) is one aligned ds_load_b64
// directly into a register pair: no VGPR repacking, pipelineable DS loads.
// ---------------------------------------------------------------------------
__global__ __launch_bounds__(256) void attn_score_gemm(
    const float* __restrict__ enc, const float* __restrict__ Uw,
    const float* __restrict__ q,  const float* __restrict__ Vw,
    const float* __restrict__ pi, const int* __restrict__ ip,
    float* __restrict__ score)
{
  constexpr int TM = 64, TN = 128, KC = 32, KP = KC + 2;  // KP even: b64 aligned
  __shared__ float As[TM][KP];   // [m][k]
  __shared__ float Bs[TN][KP];   // [n][k]  (B = U_w^T, i.e. U_w rows)

  const int tid   = threadIdx.x;
  const int wave  = tid >> 5;
  const int lane  = tid & 31;
  const int half  = lane >> 4;        // 0: lanes 0-15, 1: lanes 16-31
  const int l16   = lane & 15;
  const int waveM = wave >> 2;        // 0..1 -> M offset waveM*32
  const int waveN = wave & 3;         // 0..3 -> N offset waveN*32
  const int m0 = blockIdx.y * TM;
  const int n0 = blockIdx.x * TN;
  const int output_pos = ip[0] + 1;

  v8f c00 = {0.f,0.f,0.f,0.f,0.f,0.f,0.f,0.f};
  v8f c01 = c00, c10 = c00, c11 = c00;

  const int lm = tid >> 3;            // 0..31 (row group being loaded)
  const int lk = (tid & 7) << 2;      // 0,4,...,28 (k offset, float4)

  const int aRow = waveM * 32 + l16;  // A fragment row
  const int bRow = waveN * 32 + l16;  // B fragment row

  for (int k0 = 0; k0 < D_DIM; k0 += KC) {
    // A tile: enc rows m0..m0+63, float4 along K -> two contiguous b64 stores
    #pragma unroll
    for (int mm = 0; mm < 2; ++mm) {
      int m = mm * 32 + lm;
      float4 av = *(const float4*)(enc + (size_t)(m0 + m) * D_DIM + k0 + lk);
      *(v2f*)&As[m][lk]     = v2f{av.x, av.y};
      *(v2f*)&As[m][lk + 2] = v2f{av.z, av.w};
    }
    // B tile: U_w rows n0..n0+127 along K (coalesced), stored row-major in K
    #pragma unroll
    for (int nn = 0; nn < 4; ++nn) {
      int n = nn * 32 + lm;
      float4 bv = *(const float4*)(Uw + (size_t)(n0 + n) * D_DIM + k0 + lk);
      *(v2f*)&Bs[n][lk]     = v2f{bv.x, bv.y};
      *(v2f*)&Bs[n][lk + 2] = v2f{bv.z, bv.w};
    }
    __syncthreads();

    #pragma unroll
    for (int kk = 0; kk < KC; kk += 4) {
      // A 16x4 fragment (ISA): lanes 0-15 hold {K=kk,kk+1}, lanes 16-31
      // hold {K=kk+2,kk+3} -> one aligned b64 per fragment. B mirrored.
      const int kA = kk + half * 2;
      v2f a0 = *(const v2f*)&As[aRow][kA];
      v2f a1 = *(const v2f*)&As[aRow + 16][kA];
      v2f b0 = *(const v2f*)&Bs[bRow][kA];
      v2f b1 = *(const v2f*)&Bs[bRow + 16][kA];
      c00 = __builtin_amdgcn_wmma_f32_16x16x4_f32(false, a0, false, b0, (short)0, c00, false, false);
      c01 = __builtin_amdgcn_wmma_f32_16x16x4_f32(false, a0, false, b1, (short)0, c01, false, false);
      c10 = __builtin_amdgcn_wmma_f32_16x16x4_f32(false, a1, false, b0, (short)0, c10, false, false);
      c11 = __builtin_amdgcn_wmma_f32_16x16x4_f32(false, a1, false, b1, (short)0, c11, false, false);
    }
    __syncthreads();
  }

  // Epilogue. C VGPR r -> (M = r + half*8, N = l16) within each 16x16 tile.
  const int mBase = m0 + waveM * 32;
  const int nBase = n0 + waveN * 32;
  #pragma unroll
  for (int mi = 0; mi < 2; ++mi) {
    #pragma unroll
    for (int ni = 0; ni < 2; ++ni) {
      const v8f& cc = (mi == 0) ? (ni == 0 ? c00 : c01)
                                : (ni == 0 ? c10 : c11);
      const int d  = nBase + ni * 16 + l16;
      const float qd = q[d];
      const float vw = Vw[d];
      #pragma unroll
      for (int r = 0; r < 8; ++r) {
        const int t = mBase + mi * 16 + r + half * 8;
        float p;
        if (t < output_pos - 1) {
          p = 0.0f;
        } else {
          int idx = output_pos + T_DIM - t - 2;
          idx = idx < 0 ? 0 : (idx > T_DIM - 1 ? T_DIM - 1 : idx);
          p = pi[idx];
        }
        float val = tanhf(qd + cc[r] * p) * vw;
        #pragma unroll
        for (int off = 1; off < 16; off <<= 1)
          val += __shfl_xor(val, off, 32);  // reduce across the 16 lanes of this row
        if (l16 == 0) atomicAdd(&score[t], val);
      }
    }
  }
}

// ---------------------------------------------------------------------------
// Generic wave-per-row matvec: y[r] = act( W[r,:].x + b[r] )
// act: 0 = none, 1 = softplus
// ---------------------------------------------------------------------------
__global__ __launch_bounds__(256) void matvec_k(
    const float* __restrict__ W, const float* __restrict__ x,
    const float* __restrict__ b, float* __restrict__ y,
    int rows, int K, int act)
{
  int row = blockIdx.x * (blockDim.x >> 5) + (threadIdx.x >> 5);
  if (row >= rows) return;
  int lane = threadIdx.x & 31;
  const float4* Wr = (const float4*)(W + (size_t)row * K);
  const float4* xv = (const float4*)x;
  int n4 = K >> 2;
  float acc = 0.f;
  for (int k = lane; k < n4; k += 32) {
    float4 w = Wr[k], v = xv[k];
    acc += w.x*v.x + w.y*v.y + w.z*v.z + w.w*v.w;
  }
  for (int off = 16; off > 0; off >>= 1) acc += __shfl_xor(acc, off, 32);
  if (lane == 0) {
    float v = acc + (b ? b[row] : 0.f);
    if (act == 1) v = (v > 20.f) ? v : log1pf(expf(v));   // softplus
    y[row] = v;
  }
}

// LSTM gate matvec: g[r] = W_ih[r,:].x + b_ih[r] + W_hh[r,:].h + b_hh[r]
__global__ __launch_bounds__(256) void gates_k(
    const float* __restrict__ Wih, const float* __restrict__ Whh,
    const float* __restrict__ x,   const float* __restrict__ h,
    const float* __restrict__ bih, const float* __restrict__ bhh,
    float* __restrict__ g, int rows, int Kx, int Kh)
{
  int row = blockIdx.x * (blockDim.x >> 5) + (threadIdx.x >> 5);
  if (row >= rows) return;
  int lane = threadIdx.x & 31;
  float acc = 0.f;
  {
    const float4* Wr = (const float4*)(Wih + (size_t)row * Kx);
    const float4* xv = (const float4*)x;
    int n4 = Kx >> 2;
    for (int k = lane; k < n4; k += 32) {
      float4 w = Wr[k], v = xv[k];
      acc += w.x*v.x + w.y*v.y + w.z*v.z + w.w*v.w;
    }
  }
  {
    const float4* Wr = (const float4*)(Whh + (size_t)row * Kh);
    const float4* hv = (const float4*)h;
    int n4 = Kh >> 2;
    for (int k = lane; k < n4; k += 32) {
      float4 w = Wr[k], v = hv[k];
      acc += w.x*v.x + w.y*v.y + w.z*v.z + w.w*v.w;
    }
  }
  for (int off = 16; off > 0; off >>= 1) acc += __shfl_xor(acc, off, 32);
  if (lane == 0) g[row] = acc + bih[row] + bhh[row];
}

// Single-block softmax. If ip != nullptr, entries t < ip[0]+1 are replaced by
// 0.0 BEFORE softmax (they still participate — exactly as the reference).
__global__ __launch_bounds__(1024) void softmax_k(
    const float* __restrict__ in, float* __restrict__ out, int n,
    const int* __restrict__ ip)
{
  __shared__ float red[32];
  __shared__ float bc;
  int tid = threadIdx.x;
  int mask_below = ip ? (ip[0] + 1) : 0;

  float lmax = -INFINITY;
  for (int t = tid; t < n; t += blockDim.x) {
    float v = (t < mask_below) ? 0.f : in[t];
    lmax = fmaxf(lmax, v);
  }
  for (int off = 16; off > 0; off >>= 1) lmax = fmaxf(lmax, __shfl_xor(lmax, off, 32));
  if ((tid & 31) == 0) red[tid >> 5] = lmax;
  __syncthreads();
  if (tid < 32) {
    int nw = blockDim.x >> 5;
    float v = (tid < nw) ? red[tid] : -INFINITY;
    for (int off = 16; off > 0; off >>= 1) v = fmaxf(v, __shfl_xor(v, off, 32));
    if (tid == 0) bc = v;
  }
  __syncthreads();
  float gmax = bc;

  float lsum = 0.f;
  for (int t = tid; t < n; t += blockDim.x) {
    float v = (t < mask_below) ? 0.f : in[t];
    lsum += expf(v - gmax);
  }
  for (int off = 16; off > 0; off >>= 1) lsum += __shfl_xor(lsum, off, 32);
  __syncthreads();
  if ((tid & 31) == 0) red[tid >> 5] = lsum;
  __syncthreads();
  if (tid < 32) {
    int nw = blockDim.x >> 5;
    float v = (tid < nw) ? red[tid] : 0.f;
    for (int off = 16; off > 0; off >>= 1) v += __shfl_xor(v, off, 32);
    if (tid == 0) bc = v;
  }
  __syncthreads();
  float inv = 1.f / bc;
  for (int t = tid; t < n; t += blockDim.x) {
    float v = (t < mask_below) ? 0.f : in[t];
    out[t] = expf(v - gmax) * inv;
  }
}

// context[d] = sum_t attn[t]*enc[t,d]; also builds x = [context, data]
__global__ __launch_bounds__(256) void context_k(
    const float* __restrict__ attn, const float* __restrict__ enc,
    const float* __restrict__ data, float* __restrict__ x)
{
  int d = blockIdx.x * blockDim.x + threadIdx.x;
  if (d < D_DIM) {
    float acc = 0.f;
    for (int t = 0; t < T_DIM; ++t)
      acc += attn[t] * enc[(size_t)t * D_DIM + d];   // coalesced along d
    x[d] = acc;
  }
  if (d < IN_DIM) x[D_DIM + d] = data[d];
}

__global__ __launch_bounds__(256) void lstm_k(
    const float* __restrict__ g, const float* __restrict__ c0,
    float* __restrict__ hws, float* __restrict__ cws,
    float* __restrict__ out_h, float* __restrict__ out_c)
{
  int d = blockIdx.x * blockDim.x + threadIdx.x;
  if (d >= D_DIM) return;
  float gi = g[d], gf = g[D_DIM + d], gg = g[2*D_DIM + d], go = g[3*D_DIM + d];
  float si = 1.f / (1.f + expf(-gi));
  float sf = 1.f / (1.f + expf(-gf));
  float so = 1.f / (1.f + expf(-go));
  float cn = sf * c0[d] + si * tanhf(gg);
  float hn = so * tanhf(cn);
  hws[d] = hn; cws[d] = cn; out_h[d] = hn; out_c[d] = cn;
}

__global__ void zero_k(float* p, int n)
{
  int i = blockIdx.x * blockDim.x + threadIdx.x;
  if (i < n) p[i] = 0.f;
}

// ---------------------------------------------------------------------------
extern "C" void kernel_launch(void* const* d_in, const int* in_sizes, int n_in,
                              void* d_out, int out_size, void* d_ws, size_t ws_size,
                              hipStream_t stream)
{
  (void)in_sizes; (void)n_in; (void)out_size; (void)ws_size;

  const float* data   = (const float*)d_in[0];
  const float* h0     = (const float*)d_in[1];
  const float* c0     = (const float*)d_in[2];
  const float* enc    = (const float*)d_in[3];
  const float* pi     = (const float*)d_in[4];
  const float* W_w    = (const float*)d_in[5];
  const float* U_w    = (const float*)d_in[6];
  const float* V_w    = (const float*)d_in[7];
  const float* lin_w  = (const float*)d_in[8];
  const float* lin_b  = (const float*)d_in[9];
  const float* lin1_w = (const float*)d_in[10];
  const float* lin1_b = (const float*)d_in[11];
  const float* lin2_w = (const float*)d_in[12];
  const float* lin2_b = (const float*)d_in[13];
  const float* dense_w= (const float*)d_in[14];
  const float* dense_b= (const float*)d_in[15];
  const float* W_ih   = (const float*)d_in[16];
  const float* W_hh   = (const float*)d_in[17];
  const float* b_ih   = (const float*)d_in[18];
  const float* b_hh   = (const float*)d_in[19];
  const int*   ip     = (const int*)d_in[20];

  float* ws    = (float*)d_ws;
  float* q     = ws;             // 2048
  float* score = ws + 2048;      // 4096
  float* attn  = ws + 6144;      // 4096
  float* xbuf  = ws + 10240;     // 2560 (context ++ data)
  float* gates = ws + 12800;     // 8192
  float* hws   = ws + 20992;     // 2048
  float* cws   = ws + 23040;     // 2048
  float* fc    = ws + 25088;     // 2048
  float* l1    = ws + 27136;     // 2048

  float* out       = (float*)d_out;     // 8193 floats total
  float* out_gamma = out;               // 1
  float* out_beta  = out + 1;           // 2048
  float* out_delta = out + 2049;        // 2048
  float* out_h     = out + 4097;        // 2048
  float* out_c     = out + 6145;        // 2048

  // 1) score accumulator must start at zero (GEMM epilogue uses atomics)
  zero_k<<<(T_DIM + 255) / 256, 256, 0, stream>>>(score, T_DIM);

  // 2) q = W_w @ h0
  matvec_k<<<D_DIM / 8, 256, 0, stream>>>(W_w, h0, nullptr, q, D_DIM, D_DIM, 0);

  // 3) big fp32-WMMA GEMM fused with tanh/V_w reduction into score
  dim3 grid(D_DIM / 128, T_DIM / 64);
  attn_score_gemm<<<grid, 256, 0, stream>>>(enc, U_w, q, V_w, pi, ip, score);

  // 4) attn = softmax(mask(score))
  softmax_k<<<1, 1024, 0, stream>>>(score, attn, T_DIM, ip);

  // 5) context = attn @ enc ; x = [context, data]
  context_k<<<(D_DIM + 255) / 256, 256, 0, stream>>>(attn, enc, data, xbuf);

  // 6) LSTM gates + cell
  gates_k<<<(4 * D_DIM) / 8, 256, 0, stream>>>(W_ih, W_hh, xbuf, h0,
                                               b_ih, b_hh, gates,
                                               4 * D_DIM, IN_DIM + D_DIM, D_DIM);
  lstm_k<<<(D_DIM + 255) / 256, 256, 0, stream>>>(gates, c0, hws, cws, out_h, out_c);

  // 7) heads
  matvec_k<<<D_DIM / 8, 256, 0, stream>>>(lin_w,  hws, lin_b,  fc,       D_DIM, D_DIM, 0);
  matvec_k<<<D_DIM / 8, 256, 0, stream>>>(lin1_w, fc,  lin1_b, l1,       D_DIM, D_DIM, 0);
  matvec_k<<<D_DIM / 8, 256, 0, stream>>>(lin2_w, fc,  lin2_b, out_beta, D_DIM, D_DIM, 1); // softplus
  softmax_k<<<1, 1024, 0, stream>>>(l1, out_delta, D_DIM, nullptr);
  matvec_k<<<1, 256, 0, stream>>>(dense_w, l1, dense_b, out_gamma, 1, D_DIM, 0);
}